// policy_35742717837418
// MI455X (gfx1250) — compile-verified
//
#include <hip/hip_runtime.h>
#include <math.h>

// gfx1250 wave32 WMMA operand types for V_WMMA_F32_16X16X4_F32:
//   A: 16x4 f32  -> 64 elems / 32 lanes = 2 VGPRs
//   B: 4x16 f32  -> 2 VGPRs
//   C/D: 16x16 f32 -> 8 VGPRs
typedef __attribute__((ext_vector_type(2))) float v2f;
typedef __attribute__((ext_vector_type(8))) float v8f;

__launch_bounds__(128)
__global__ void policy_fused_kernel(const float* __restrict__ s,   // [1,4,25,50]
                                    const float* __restrict__ w1,  // [8,4,1,3]
                                    const float* __restrict__ b1,  // [8]
                                    const float* __restrict__ w2,  // [2,8,1,48]
                                    const float* __restrict__ b2,  // [2]
                                    const float* __restrict__ wf,  // [10,50]
                                    const float* __restrict__ bf,  // [10]
                                    float* __restrict__ out) {     // [10]
  __shared__ float y1[8 * 1200];  // conv1 out: [c][h*48 + w]
  __shared__ float y2[50];        // conv2 out flattened: [m*25 + h]
  __shared__ float logits[10];

  const int lane  = threadIdx.x & 31;
  const int wave  = threadIdx.x >> 5;
  const int mrow  = lane & 15;        // M row (for A lanes) / N col (for B lanes)
  const int kbase = (lane >> 4) * 2;  // K sub-offset within each K=4 step

  // ============ Stage 1: conv1 as WMMA f32 16x16x4 ============
  // D[16x16] = A[16x4] * B[4x16] + C, M: out-channel (8 real, 8 zero-pad),
  // K: 4 in-ch x 3 taps = 12 (3 steps), N: 25*48 = 1200 positions (75 tiles).
  // A (weights) is loop-invariant: hoist all 3 K-steps.
  v2f a1[3];
  #pragma unroll
  for (int t = 0; t < 3; ++t) {
    const int kk = t * 4 + kbase;
    a1[t].x = (mrow < 8) ? w1[mrow * 12 + kk]     : 0.0f;
    a1[t].y = (mrow < 8) ? w1[mrow * 12 + kk + 1] : 0.0f;
  }
  for (int T = wave; T < 75; T += 4) {            // wave-uniform tiling
    const int p = T * 16 + mrow;                  // output position (N index)
    const int h = p / 48;
    const int w = p - h * 48;
    v8f acc = {};
    #pragma unroll
    for (int t = 0; t < 3; ++t) {
      const int kk0 = t * 4 + kbase;
      const int c0 = kk0 / 3, tap0 = kk0 - c0 * 3;
      const int kk1 = kk0 + 1;
      const int c1 = kk1 / 3, tap1 = kk1 - c1 * 3;
      v2f b;
      b.x = s[c0 * 1250 + h * 50 + w + tap0];     // w+tap <= 49 < 50: in-range
      b.y = s[c1 * 1250 + h * 50 + w + tap1];
      acc = __builtin_amdgcn_wmma_f32_16x16x4_f32(false, a1[t], false, b,
                                                  (short)0, acc, false, false);
    }
    // D layout: VGPR r = row M=r on lanes 0..15 (rows 8..15 on lanes 16..31 are pad)
    if (lane < 16) {
      #pragma unroll
      for (int r = 0; r < 8; ++r) {
        const float v = acc[r] + b1[r];
        y1[r * 1200 + T * 16 + lane] = (v > 0.0f) ? v : 0.01f * v;  // leaky relu
      }
    }
  }
  __syncthreads();

  // ============ Stage 2: conv2 as WMMA f32 16x16x4 ============
  // M: 2 out-channels (pad 16), K: 8 ch x 48 taps = 384 (96 steps),
  // N: 25 rows (2 tiles of 16, tail lanes zero-padded). Waves 0,1 take one tile each.
  if (wave < 2) {
    const int T = wave;
    const int h = T * 16 + mrow;                  // N index = output row h
    v8f acc = {};
    for (int t = 0; t < 96; ++t) {
      const int kk0 = t * 4 + kbase;
      const int kk1 = kk0 + 1;
      v2f a, b;
      a.x = (mrow < 2) ? w2[mrow * 384 + kk0] : 0.0f;
      a.y = (mrow < 2) ? w2[mrow * 384 + kk1] : 0.0f;
      const int c0 = kk0 / 48, kt0 = kk0 - c0 * 48;
      const int c1 = kk1 / 48, kt1 = kk1 - c1 * 48;
      b.x = (h < 25) ? y1[c0 * 1200 + h * 48 + kt0] : 0.0f;
      b.y = (h < 25) ? y1[c1 * 1200 + h * 48 + kt1] : 0.0f;
      acc = __builtin_amdgcn_wmma_f32_16x16x4_f32(false, a, false, b,
                                                  (short)0, acc, false, false);
    }
    if (lane < 16) {
      const int hh = T * 16 + lane;
      if (hh < 25) {
        const float v0 = acc[0] + b2[0];
        const float v1 = acc[1] + b2[1];
        y2[hh]      = (v0 > 0.0f) ? v0 : 0.01f * v0;  // reshape(50): m*25+h
        y2[25 + hh] = (v1 > 0.0f) ? v1 : 0.01f * v1;
      }
    }
  }
  __syncthreads();

  // ============ Stage 3: FC [10,50] + softmax ============
  if (threadIdx.x < 10) {
    float acc = bf[threadIdx.x];
    #pragma unroll 5
    for (int i = 0; i < 50; ++i) acc += wf[threadIdx.x * 50 + i] * y2[i];
    logits[threadIdx.x] = acc;
  }
  __syncthreads();
  if (threadIdx.x < 10) {
    float m = logits[0];
    #pragma unroll
    for (int j = 1; j < 10; ++j) m = fmaxf(m, logits[j]);
    float sum = 0.0f;
    #pragma unroll
    for (int j = 0; j < 10; ++j) sum += expf(logits[j] - m);
    out[threadIdx.x] = expf(logits[threadIdx.x] - m) / sum;
  }
}

extern "C" void kernel_launch(void* const* d_in, const int* in_sizes, int n_in,
                              void* d_out, int out_size, void* d_ws, size_t ws_size,
                              hipStream_t stream) {
  (void)in_sizes; (void)n_in; (void)out_size; (void)d_ws; (void)ws_size;
  policy_fused_kernel<<<1, 128, 0, stream>>>(
      (const float*)d_in[0],   // s  [1,4,25,50]
      (const float*)d_in[1],   // w1 [8,4,1,3]
      (const float*)d_in[2],   // b1 [8]
      (const float*)d_in[3],   // w2 [2,8,1,48]
      (const float*)d_in[4],   // b2 [2]
      (const float*)d_in[5],   // wf [10,50]
      (const float*)d_in[6],   // bf [10]
      (float*)d_out);          // [10]
}